// Probabilistic_DAG_Generator_From_Roots_17806934409652
// MI455X (gfx1250) — compile-verified
//
#include <hip/hip_runtime.h>
#include <hip/hip_bf16.h>

// Probabilistic DAG generator (forward value of straight-through gumbel BFS).
// N = 1024 nodes. dag[i,j] = 1.0 iff node i was dequeued by the BFS and
// hard_e[i,j] & ~root_hard[j] & ~anc_at_dequeue(i)[j].
//
// MI455X design:
//  - prep kernel: memory-bound element-wise compares -> bitmasks, zero d_out.
//  - bfs kernel : ONE workgroup, 1024 threads (32 waves, wave32), ~264 KB LDS
//    (legal only on CDNA5: 320 KB/WGP). hard_e bitmask staged into LDS via
//    global_load_async_to_lds_b128 (ASYNCcnt), anc bitmask lives in LDS with
//    a 33-word row stride (gcd(33,64)=1 -> bank-conflict-free per-row access
//    across the 64 LDS banks), per-step masks via wave32 ballot, ordered FIFO
//    enqueue by thread 0.

#define NNODES 1024
#define NW     32            // 32-bit words per 1024-bit row
#define ANCS   33            // padded anc row stride (words) -> conflict-free

__device__ __forceinline__ unsigned ballot32(bool p) {
  return __builtin_amdgcn_ballot_w32(p);
}

// ---------------------------------------------------------------------------
// Kernel 1: hard gumbel bitmasks + zero the output. 1 thread per (i,j).
// ---------------------------------------------------------------------------
__global__ __launch_bounds__(256) void dag_prep_kernel(
    const float* __restrict__ root_probs,   // (N)
    const float* __restrict__ edge_probs,   // (N,N)
    const float* __restrict__ g_roots,      // (N,2)
    const float* __restrict__ g_edges,      // (N,N,2)
    unsigned* __restrict__ hardE,           // (N*N/32) bitmask
    unsigned* __restrict__ rootH,           // (N/32)   bitmask
    float* __restrict__ dag)                // (N,N) -> zeroed here
{
  const int gid  = blockIdx.x * 256 + threadIdx.x;   // 0 .. N*N-1
  const int lane = threadIdx.x & 31;

  dag[gid] = 0.0f;

  const float  p = edge_probs[gid];
  const float2 g = ((const float2*)g_edges)[gid];    // (g0, g1), coalesced 8B
  // z0 = p + g0 ; z1 = (1 - p) + g1 ; hard = z0 > z1   (same op order as ref)
  const bool hard = (p + g.x) > ((1.0f - p) + g.y);
  const unsigned m = ballot32(hard);
  if (lane == 0) hardE[gid >> 5] = m;

  if (gid < NNODES) {                                 // warp-uniform (1024%256==0)
    const float  rp = root_probs[gid];
    const float2 gr = ((const float2*)g_roots)[gid];
    const bool rh = (rp + gr.x) > ((1.0f - rp) + gr.y);
    const unsigned rm = ballot32(rh);
    if (lane == 0) rootH[gid >> 5] = rm;
  }
}

// ---------------------------------------------------------------------------
// Kernel 2: sequential FIFO BFS, single workgroup, everything LDS-resident.
// Dynamic LDS layout (u32 words):
//   [0      , 33792) anc bitmask   (33-word padded rows)         132 KB
//   [33792  , 66560) hardE bitmask (staged via async DMA)        128 KB
//   [66560  , 67584) queue (node indices)                          4 KB
//   [67584  , 67616) newmask (per-wave ballot words)
//   [67616  , 67624) ctrl: [0]=head [1]=tail
// ---------------------------------------------------------------------------
#define LDS_WORDS 67624

__global__ __launch_bounds__(1024) void dag_bfs_kernel(
    const unsigned* __restrict__ hardE_g,
    const unsigned* __restrict__ rootH_g,
    float* __restrict__ dag)
{
  extern __shared__ unsigned smem[];
  unsigned* anc   = smem;            // 1024 * 33
  unsigned* hardE = smem + 33792;    // 32768
  unsigned* q     = smem + 66560;    // 1024
  unsigned* nm    = smem + 67584;    // 32
  unsigned* ctrl  = smem + 67616;    // 8

  const int t    = threadIdx.x;      // == j, 0..1023
  const int w    = t >> 5;           // wave id == word index for bit j
  const int lane = t & 31;

  // ---- Stage hardE (128 KB) global -> LDS with CDNA5 async DMA (ASYNCcnt).
  // 8192 x 16B chunks; thread t copies chunks t, t+1024, ..., t+7*1024.
  #pragma unroll
  for (int k = 0; k < 8; ++k) {
    const unsigned c = (unsigned)(k * 1024 + t);
    unsigned lds_off = (unsigned)(unsigned long long)(const void*)&hardE[c * 4u];
    const unsigned* gp = hardE_g + (size_t)c * 4u;
    asm volatile("global_load_async_to_lds_b128 %0, %1, off"
                 :: "v"(lds_off), "v"(gp) : "memory");
  }

  // ---- anc := I (each thread owns its padded row), overlapped with the DMA.
  #pragma unroll
  for (int w2 = 0; w2 < NW; ++w2)
    anc[t * ANCS + w2] = (w2 == w) ? (1u << lane) : 0u;

  // ---- per-thread constants: root bit / enqueued-once bit.
  const bool rbit = (rootH_g[w] >> lane) & 1u;
  bool inq = rbit;

  // ---- thread 0: initial queue = roots in ascending order.
  if (t == 0) {
    unsigned tail = 0;
    for (int w2 = 0; w2 < NW; ++w2) {
      unsigned m = rootH_g[w2];
      while (m) {
        const int b = __builtin_ctz(m);
        q[tail++] = (unsigned)(w2 * 32 + b);
        m &= m - 1;
      }
    }
    ctrl[0] = 0;       // head
    ctrl[1] = tail;    // tail
  }

  asm volatile("s_wait_asynccnt 0" ::: "memory");   // this wave's DMA done

  // ---- BFS: at most N active steps (each node dequeued at most once).
  for (int step = 0; step < NNODES; ++step) {
    __syncthreads();                      // publish ctrl/queue/anc updates
    const unsigned head = ctrl[0];
    const unsigned tail = ctrl[1];
    if (head >= tail) break;              // uniform: remaining dag rows stay 0

    const unsigned i = q[head];

    const unsigned anc_iw  = anc[i * ANCS + w];    // same-word wave broadcast
    const unsigned hard_iw = hardE[i * NW + w];    // same-word wave broadcast
    const bool anc_b  = (anc_iw  >> lane) & 1u;
    const bool hard_b = (hard_iw >> lane) & 1u;
    const bool hit    = hard_b && !rbit && !anc_b; // never true for j == i

    dag[(size_t)i * NNODES + t] = hit ? 1.0f : 0.0f;   // coalesced row store

    const bool isnew = hit && !inq;
    inq = inq || isnew;
    const unsigned bm = ballot32(isnew);
    if (lane == 0) nm[w] = bm;            // fresh every step, no clear needed

    // anc[j] |= anc[i] for every hit j (row i itself is never written).
    // Reads of row i broadcast; writes to own padded row are conflict-free.
    if (hit) {
      #pragma unroll
      for (int w2 = 0; w2 < NW; ++w2)
        anc[t * ANCS + w2] |= anc[i * ANCS + w2];
    }

    __syncthreads();                      // nm[] complete before enqueue

    if (t == 0) {                         // ordered FIFO enqueue (ascending j)
      unsigned tl = tail;
      for (int w2 = 0; w2 < NW; ++w2) {
        unsigned m = nm[w2];
        while (m) {
          const int b = __builtin_ctz(m);
          q[tl++] = (unsigned)(w2 * 32 + b);
          m &= m - 1;
        }
      }
      ctrl[0] = head + 1;
      ctrl[1] = tl;
    }
  }
}

// ---------------------------------------------------------------------------
extern "C" void kernel_launch(void* const* d_in, const int* in_sizes, int n_in,
                              void* d_out, int out_size, void* d_ws, size_t ws_size,
                              hipStream_t stream) {
  const float* root_probs = (const float*)d_in[0];   // (N)
  const float* edge_probs = (const float*)d_in[1];   // (N,N)
  const float* g_roots    = (const float*)d_in[2];   // (N,2)
  const float* g_edges    = (const float*)d_in[3];   // (N,N,2)
  float* dag = (float*)d_out;

  unsigned* hardE = (unsigned*)d_ws;                     // 128 KB
  unsigned* rootH = hardE + (NNODES * NNODES / 32);      // 128 B

  dag_prep_kernel<<<(NNODES * NNODES) / 256, 256, 0, stream>>>(
      root_probs, edge_probs, g_roots, g_edges, hardE, rootH, dag);

  const size_t lds_bytes = (size_t)LDS_WORDS * sizeof(unsigned);  // ~264 KB
  (void)hipFuncSetAttribute((const void*)dag_bfs_kernel,
                            hipFuncAttributeMaxDynamicSharedMemorySize,
                            (int)lds_bytes);
  dag_bfs_kernel<<<1, 1024, lds_bytes, stream>>>(hardE, rootH, dag);
}